// GCN_39427799777280
// MI455X (gfx1250) — compile-verified
//
#include <hip/hip_runtime.h>
#include <math.h>

typedef float v2f __attribute__((ext_vector_type(2)));
typedef float v8f __attribute__((ext_vector_type(8)));

// ---------------------------------------------------------------------------
// deg[i] = 1.0f  (self-loop contribution)
// ---------------------------------------------------------------------------
__global__ void k_fill_one(float* __restrict__ deg, int n) {
    int i = blockIdx.x * blockDim.x + threadIdx.x;
    if (i < n) deg[i] = 1.0f;
}

// deg[dst[e]] += 1  (unit edge weights)
__global__ void k_accum_deg(const int* __restrict__ dst, float* __restrict__ deg, int E) {
    int e = blockIdx.x * blockDim.x + threadIdx.x;
    if (e < E) unsafeAtomicAdd(&deg[dst[e]], 1.0f);
}

// deg -> rsqrt(deg) in place (deg >= 1 always, self-loop)
__global__ void k_rsqrt_inplace(float* __restrict__ deg, int n) {
    int i = blockIdx.x * blockDim.x + threadIdx.x;
    if (i < n) {
        float d = deg[i];
        deg[i] = d > 0.0f ? rsqrtf(d) : 0.0f;
    }
}

// ---------------------------------------------------------------------------
// CDNA5 async cache->LDS copy (no VGPR round-trip), tracked with ASYNCcnt.
// ldsoff: byte offset within the workgroup's LDS allocation (addrspace(3)
// pointer value); g: 64-bit global address.
// ---------------------------------------------------------------------------
__device__ __forceinline__ void async_g2l_b32(unsigned int ldsoff, const void* g) {
    asm volatile("global_load_async_to_lds_b32 %0, %1, off"
                 :: "v"(ldsoff), "v"(g)
                 : "memory");
}
__device__ __forceinline__ void wait_asynccnt0() {
    asm volatile("s_wait_asynccnt 0" ::: "memory");
}

// ---------------------------------------------------------------------------
// C[M,N] = (RELU ? relu(A) : A)[M,K] @ B[K,N]   via V_WMMA_F32_16X16X4_F32
//
// Block = 128 threads = 4 waves. The block's 64-wide B panel (K x 64, K<=128)
// is staged once in LDS via GLOBAL_LOAD_ASYNC_TO_LDS_B32 in a "paired-k"
// layout so the WMMA B operand (B[kk][col], B[kk+1][col]) is a single
// aligned ds_load_b64:
//    Bs[(k>>1)*LDB_ + col*2 + (k&1)],  LDB_ = 160 floats
// Row stride 160 = +32 banks (mod 64), so lanes 0-15 (row kk) and lanes
// 16-31 (row kk+2) hit disjoint bank halves -> conflict-free.
// Each wave computes a 16x64 output tile (4 v8f accumulators).
// Inner loop is branch-free: OOB columns are clamped for loads (their
// results land in discarded output columns) and guarded only at the store.
// ---------------------------------------------------------------------------
#define LDB_ 160

template <int RELU>
__global__ void k_gemm_wmma_f32(const float* __restrict__ A,
                                const float* __restrict__ B,
                                float* __restrict__ C,
                                int M, int N, int K) {
    __shared__ float Bs[64 * LDB_];              // supports K <= 128 (40 KB)

    const int tid   = threadIdx.x;
    const int nbase = blockIdx.y * 64;

    // Cooperative async fill of the B panel (clamped columns), paired layout.
    for (int idx = tid; idx < K * 64; idx += 128) {
        const int k = idx >> 6;
        const int c = idx & 63;
        int gc = nbase + c;
        if (gc >= N) gc = N - 1;                 // garbage col -> discarded at store
        const unsigned int loff =
            (unsigned int)(size_t)(&Bs[(k >> 1) * LDB_ + c * 2 + (k & 1)]);
        async_g2l_b32(loff, B + (size_t)k * N + gc);
    }
    wait_asynccnt0();                            // this wave's async copies done
    __syncthreads();                             // all waves' copies visible

    const int wave  = tid >> 5;
    const int lane  = tid & 31;
    const int mtile = blockIdx.x * 4 + wave;
    if (mtile * 16 >= M) return;                 // wave-uniform, after barrier

    const int lrow  = lane & 15;
    const int khalf = lane >> 4;                 // 0 or 1

    int row = mtile * 16 + lrow;
    if (row >= M) row = M - 1;                   // load clamp (M%16==0 in practice)
    const float* Arow = A + (size_t)row * K;

    v8f acc0 = {}, acc1 = {}, acc2 = {}, acc3 = {};

#pragma unroll 2
    for (int k0 = 0; k0 < K; k0 += 4) {
        const int kk = k0 + 2 * khalf;           // even
        // A tile 16x4: lane holds (row, kk) and (row, kk+1)
        v2f a = *(const v2f*)(Arow + kk);
        if (RELU) { a.x = fmaxf(a.x, 0.0f); a.y = fmaxf(a.y, 0.0f); }

        const float* bsrow = &Bs[(kk >> 1) * LDB_];
#pragma unroll
        for (int t = 0; t < 4; ++t) {
            const v2f b = *(const v2f*)(bsrow + (t * 16 + lrow) * 2);
            switch (t) {
            case 0: acc0 = __builtin_amdgcn_wmma_f32_16x16x4_f32(false, a, false, b, (short)0, acc0, false, false); break;
            case 1: acc1 = __builtin_amdgcn_wmma_f32_16x16x4_f32(false, a, false, b, (short)0, acc1, false, false); break;
            case 2: acc2 = __builtin_amdgcn_wmma_f32_16x16x4_f32(false, a, false, b, (short)0, acc2, false, false); break;
            case 3: acc3 = __builtin_amdgcn_wmma_f32_16x16x4_f32(false, a, false, b, (short)0, acc3, false, false); break;
            }
        }
    }

    // C/D layout: VGPR j -> row (mtile*16 + j + 8*khalf), col = nbase + t*16 + lrow
    const int rbase = mtile * 16 + 8 * khalf;
#pragma unroll
    for (int t = 0; t < 4; ++t) {
        const int col = nbase + t * 16 + lrow;
        if (col >= N) continue;
        const v8f acc = (t == 0) ? acc0 : (t == 1) ? acc1 : (t == 2) ? acc2 : acc3;
#pragma unroll
        for (int j = 0; j < 8; ++j) {
            const int r = rbase + j;
            if (r < M) C[(size_t)r * N + col] = acc[j];
        }
    }
}

// ---------------------------------------------------------------------------
// agg[i,f] = h[i,f] * dinv[i]^2 + bias[f]   (self-loop message + post-agg bias)
// ---------------------------------------------------------------------------
__global__ void k_selfloop_bias(const float* __restrict__ h,
                                const float* __restrict__ dinv,
                                const float* __restrict__ bias,
                                float* __restrict__ agg, int n, int F) {
    int idx = blockIdx.x * blockDim.x + threadIdx.x;
    if (idx < n * F) {
        int i = idx / F, f = idx - i * F;
        float di = dinv[i];
        agg[idx] = h[idx] * di * di + bias[f];
    }
}

// ---------------------------------------------------------------------------
// Edge scatter: one wave per edge. lane moves one float4 (128 feats / 32 lanes)
// agg[dst] += h[src] * dinv[src]*dinv[dst]
// ---------------------------------------------------------------------------
__global__ void k_scatter_edges(const float* __restrict__ h,
                                const int* __restrict__ src,
                                const int* __restrict__ dst,
                                const float* __restrict__ dinv,
                                float* __restrict__ agg, int E) {
    const int wid  = (blockIdx.x * blockDim.x + threadIdx.x) >> 5;
    const int lane = threadIdx.x & 31;
    if (wid >= E) return;                        // wave-uniform
    const int s = src[wid];
    const int d = dst[wid];
    const float nrm = dinv[s] * dinv[d];
    const float4 v = ((const float4*)(h + (size_t)s * 128))[lane];
    float* op = agg + (size_t)d * 128 + lane * 4;
    unsafeAtomicAdd(op + 0, v.x * nrm);
    unsafeAtomicAdd(op + 1, v.y * nrm);
    unsafeAtomicAdd(op + 2, v.z * nrm);
    unsafeAtomicAdd(op + 3, v.w * nrm);
}

// ---------------------------------------------------------------------------
// Row-wise log_softmax over 40 columns; bias bl added pre-softmax.
// One wave per row; wave32 shuffle reductions.
// ---------------------------------------------------------------------------
__global__ void k_logsoftmax(const float* __restrict__ logits,
                             const float* __restrict__ bl,
                             float* __restrict__ out, int n, int C) {
    const int row  = (blockIdx.x * blockDim.x + threadIdx.x) >> 5;
    const int lane = threadIdx.x & 31;
    if (row >= n) return;                        // wave-uniform

    const float NEG = -3.4e38f;
    float x0 = (lane      < C) ? logits[(size_t)row * C + lane]       + bl[lane]      : NEG;
    float x1 = (lane + 32 < C) ? logits[(size_t)row * C + lane + 32]  + bl[lane + 32] : NEG;

    float m = fmaxf(x0, x1);
#pragma unroll
    for (int off = 16; off > 0; off >>= 1) m = fmaxf(m, __shfl_xor(m, off, 32));

    float s = 0.0f;
    if (lane      < C) s += expf(x0 - m);
    if (lane + 32 < C) s += expf(x1 - m);
#pragma unroll
    for (int off = 16; off > 0; off >>= 1) s += __shfl_xor(s, off, 32);

    const float lse = logf(s);
    if (lane      < C) out[(size_t)row * C + lane]      = x0 - m - lse;
    if (lane + 32 < C) out[(size_t)row * C + lane + 32] = x1 - m - lse;
}

// ---------------------------------------------------------------------------
extern "C" void kernel_launch(void* const* d_in, const int* in_sizes, int n_in,
                              void* d_out, int out_size, void* d_ws, size_t ws_size,
                              hipStream_t stream) {
    const int IN = 128, HID = 128, OUTC = 40;
    const int Nn = in_sizes[0] / IN;       // 50000
    const int E  = in_sizes[1] / 2;        // 600000

    const float* x  = (const float*)d_in[0];
    const int*   src= (const int*)d_in[1];
    const int*   dst= src + E;
    const float* W1 = (const float*)d_in[2];
    const float* b1 = (const float*)d_in[3];
    const float* W2 = (const float*)d_in[4];
    const float* b2 = (const float*)d_in[5];
    const float* Wl = (const float*)d_in[6];
    const float* bl = (const float*)d_in[7];

    float* ws   = (float*)d_ws;
    float* dinv = ws;                      // deg -> rsqrt in place
    float* bufA = ws + Nn;                 // Nn*128
    float* bufB = bufA + (size_t)Nn * HID; // Nn*128

    float* out_ls = (float*)d_out;                    // [Nn, 40]
    float* x_emb  = out_ls + (size_t)Nn * OUTC;       // [Nn, 128]

    const int Mt = (Nn + 15) / 16;                    // 3125
    dim3 gemmBlk(128);
    dim3 gemmGrdH((Mt + 3) / 4, HID / 64);            // N=128 -> y=2
    dim3 gemmGrdO((Mt + 3) / 4, (OUTC + 63) / 64);    // N=40  -> y=1

    // degrees + symmetric norm
    k_fill_one      <<<(Nn + 255) / 256, 256, 0, stream>>>(dinv, Nn);
    k_accum_deg     <<<(E  + 255) / 256, 256, 0, stream>>>(dst, dinv, E);
    k_rsqrt_inplace <<<(Nn + 255) / 256, 256, 0, stream>>>(dinv, Nn);

    // layer 1: h = x @ W1 ; agg1 = scatter(norm * h) + selfloop + b1
    k_gemm_wmma_f32<0> <<<gemmGrdH, gemmBlk, 0, stream>>>(x, W1, bufA, Nn, HID, IN);
    k_selfloop_bias <<<((size_t)Nn * HID + 255) / 256, 256, 0, stream>>>(bufA, dinv, b1, bufB, Nn, HID);
    k_scatter_edges <<<(E + 7) / 8, 256, 0, stream>>>(bufA, src, dst, dinv, bufB, E);

    // layer 2: h2 = relu(agg1) @ W2 ; x_emb = scatter(norm * h2) + selfloop + b2
    k_gemm_wmma_f32<1> <<<gemmGrdH, gemmBlk, 0, stream>>>(bufB, W2, bufA, Nn, HID, HID);
    k_selfloop_bias <<<((size_t)Nn * HID + 255) / 256, 256, 0, stream>>>(bufA, dinv, b2, x_emb, Nn, HID);
    k_scatter_edges <<<(E + 7) / 8, 256, 0, stream>>>(bufA, src, dst, dinv, x_emb, E);

    // classifier: logits = x_emb @ Wl (+bl in softmax kernel) ; log_softmax
    k_gemm_wmma_f32<0> <<<gemmGrdO, gemmBlk, 0, stream>>>(x_emb, Wl, bufB, Nn, OUTC, HID);
    k_logsoftmax    <<<(Nn + 7) / 8, 256, 0, stream>>>(bufB, bl, out_ls, Nn, OUTC);
}